// CharacterAwareEncoder_55542517072520
// MI455X (gfx1250) — compile-verified
//
#include <hip/hip_runtime.h>
#include <hip/hip_bf16.h>

// CharacterAwareEncoder: out[b,s,0:412] = table[ids[b,s]], out[b,s,412:512] =
// sin((j/1000)*id) masked by id>0.  Pure bandwidth problem (~118MB traffic,
// ~5-10us at 23.3 TB/s).  No matmul -> no WMMA; the CDNA5 path here is the
// async global<->LDS DMA pipe (ASYNCcnt-tracked) for the gather-copy.

#define EMB_F      412          // embedding floats per token
#define OUT_F      512          // output floats per token
#define EMB_CHUNKS 103          // 412 / 4 float4 chunks (1648B, 16B aligned)
#define OUT_CHUNKS 128          // 512 / 4
#define BLOCK      128          // 4 waves: lanes 0..102 copy, 103..127 sin

typedef int v4i __attribute__((ext_vector_type(4)));
typedef __attribute__((address_space(1))) v4i gv4i;   // global int4
typedef __attribute__((address_space(3))) v4i lv4i;   // LDS int4

#if __has_builtin(__builtin_amdgcn_global_load_async_to_lds_b128) && \
    __has_builtin(__builtin_amdgcn_global_store_async_from_lds_b128)
#define USE_ASYNC_LDS 1
#else
#define USE_ASYNC_LDS 0
#endif

__global__ __launch_bounds__(BLOCK) void char_enc_kernel(
    const int* __restrict__ ids,
    const float* __restrict__ table,
    float* __restrict__ out,
    int n_tokens)
{
    const int token = blockIdx.x;
    if (token >= n_tokens) return;
    const int tid = threadIdx.x;

    const int id = ids[token];   // block-uniform -> scalar load

    float4* __restrict__ out4 =
        reinterpret_cast<float4*>(out) + (size_t)token * OUT_CHUNKS;

#if USE_ASYNC_LDS
    __shared__ float4 stage[EMB_CHUNKS];
#endif

    if (tid < EMB_CHUNKS) {
        const float4* src =
            reinterpret_cast<const float4*>(table + (size_t)id * EMB_F) + tid;
#if USE_ASYNC_LDS
        gv4i* gsrc = (gv4i*)src;              // strip const + cast to AS(1) int4*
        lv4i* lptr = (lv4i*)&stage[tid];      // AS(3) int4*
        // global -> LDS, 16B per active lane, tracked by ASYNCcnt
        __builtin_amdgcn_global_load_async_to_lds_b128(gsrc, lptr, 0, 0);
        // each lane re-reads only the LDS bytes it loaded; per-wave wait on
        // its single outstanding async load is sufficient (no barrier).
#if __has_builtin(__builtin_amdgcn_s_wait_asynccnt)
        __builtin_amdgcn_s_wait_asynccnt(0);
#else
        asm volatile("s_wait_asynccnt 0x0" ::: "memory");
#endif
        gv4i* gdst = (gv4i*)(out4 + tid);
        // LDS -> global; S_ENDPGM's implicit wait-idle drains these stores
        __builtin_amdgcn_global_store_async_from_lds_b128(gdst, lptr, 0, 0);
#else
        out4[tid] = *src;   // fallback: coalesced B128 load/store via VGPRs
#endif
    } else {
        // lanes 103..127: 25 float4 = 100 sinusoid char features
        const int j0 = (tid - EMB_CHUNKS) * 4;           // 0,4,...,96
        float4 v = make_float4(0.f, 0.f, 0.f, 0.f);
        if (id > 0) {
            const float fid = (float)id * 0.001f;        // id/1000
            v.x = __sinf(fid * (float)(j0 + 0));
            v.y = __sinf(fid * (float)(j0 + 1));
            v.z = __sinf(fid * (float)(j0 + 2));
            v.w = __sinf(fid * (float)(j0 + 3));
        }
        out4[tid] = v;   // 103 + (tid-103) == tid: row stays fully coalesced
    }
}

extern "C" void kernel_launch(void* const* d_in, const int* in_sizes, int n_in,
                              void* d_out, int out_size, void* d_ws, size_t ws_size,
                              hipStream_t stream) {
    const int*   ids   = (const int*)d_in[0];    // word_ids   [16,2048]
    const float* table = (const float*)d_in[1];  // emb table  [32000,412]
    float*       out   = (float*)d_out;          // [16,2048,512]

    const int n_tokens = out_size / OUT_F;       // 32768

    hipLaunchKernelGGL(char_enc_kernel,
                       dim3(n_tokens), dim3(BLOCK), 0, stream,
                       ids, table, out, n_tokens);
}